// SelectiveScanState_5136780886274
// MI455X (gfx1250) — compile-verified
//
#include <hip/hip_runtime.h>
#include <hip/hip_bf16.h>

// Selective scan (Mamba SSM), shapes: B=2, D=1024, N=16, L=2048.
// h_t = exp(delta_t * A[d,n]) * h_{t-1} + (delta_t * u_t) * B[b,n,t]
// out[b,n,d,l] = h_l  (f32, 256 MB output -> store-bandwidth bound on MI455X)
//
// Blocking: one 16-wave block per (b,d); wave w = state n; chunk = 128
// timesteps; each lane owns 4 serial timesteps (16B vector I/O), wave
// shuffle-scan only over 32 lane-aggregates. Async global->LDS staging.

#define BQ 2
#define DM 1024
#define NS 16
#define LQ 2048
#define TPL 4              // timesteps per lane
#define CH (32 * TPL)      // 128 timesteps per chunk
#define NCH (LQ / CH)      // 16 chunks

typedef int   v4i __attribute__((ext_vector_type(4)));
typedef float v4f __attribute__((ext_vector_type(4)));

// ---- CDNA5 async global->LDS copy (ASYNCcnt path) -------------------------
__device__ __forceinline__ void async_copy16(const void* gsrc, void* ldst) {
#if __has_builtin(__builtin_amdgcn_global_load_async_to_lds_b128)
  typedef __attribute__((address_space(1))) v4i gv4i;
  typedef __attribute__((address_space(3))) v4i lv4i;
  __builtin_amdgcn_global_load_async_to_lds_b128((gv4i*)gsrc, (lv4i*)ldst,
                                                 /*offset=*/0, /*cpol=*/0);
#else
  unsigned lds_off = (unsigned)(unsigned long long)ldst;  // low 32 bits = LDS address
  asm volatile("global_load_async_to_lds_b128 %0, %1, off"
               :: "v"(lds_off), "v"(gsrc)
               : "memory");
#endif
}

__device__ __forceinline__ void wait_async0() {
#if __has_builtin(__builtin_amdgcn_s_wait_asynccnt)
  __builtin_amdgcn_s_wait_asynccnt(0);
#else
  asm volatile("s_wait_asynccnt 0x0" ::: "memory");
#endif
}

// ---------------------------------------------------------------------------
__global__ __launch_bounds__(NS * 32, 1)
void SelectiveScanState_kernel(const float* __restrict__ u,
                               const float* __restrict__ delta,
                               const float* __restrict__ A,
                               const float* __restrict__ Bm,
                               float* __restrict__ out) {
  // Double-buffered staging: delta/u shared across all 16 state-waves,
  // one private B row per wave.  2*(128+128+16*128)*4B = 18 KB LDS.
  __shared__ float s_d[2][CH];
  __shared__ float s_u[2][CH];
  __shared__ float s_B[2][NS][CH];

  const int bd = blockIdx.x;        // one block per (b, d)
  const int b  = bd >> 10;          // / DM
  const int d  = bd & (DM - 1);
  const int w  = threadIdx.x >> 5;  // wave id == state index n
  const int t  = threadIdx.x & 31;  // lane; owns timesteps 4t..4t+3 of chunk

  const float A_dn = A[d * NS + w];                       // uniform per wave
  const float* dlt_base = delta + (size_t)(b * DM + d) * LQ;
  const float* u_base   = u     + (size_t)(b * DM + d) * LQ;
  const float* B_base   = Bm    + (size_t)(b * NS + w) * LQ;
  float* out_base = out + (((size_t)b * NS + w) * DM + d) * LQ;

  auto issue_chunk = [&](int c, int buf) {
    const int l0 = c * CH;
    const int o  = t * TPL;  // 16B per lane, full wave = 512B per instruction
    if (w == 0) {            // wave 0 stages the shared delta|u streams
      async_copy16(dlt_base + l0 + o, &s_d[buf][o]);
      async_copy16(u_base   + l0 + o, &s_u[buf][o]);
    }
    async_copy16(B_base + l0 + o, &s_B[buf][w][o]);  // private B row
  };

  issue_chunk(0, 0);  // prefetch chunk 0

  float h = 0.0f;     // chunk carry (state at end of previous chunk)
  for (int c = 0; c < NCH; ++c) {
    const int buf = c & 1;

    wait_async0();     // each wave waits only its own ASYNCcnt
    __syncthreads();   // make wave 0's delta/u staging visible to all waves

    if (c + 1 < NCH) issue_chunk(c + 1, buf ^ 1);  // prefetch next chunk

    // ---- per-lane data: 4 consecutive timesteps, 16B LDS reads ----
    const int o = t * TPL;
    const v4f d4 = *(const v4f*)&s_d[buf][o];
    const v4f u4 = *(const v4f*)&s_u[buf][o];
    const v4f b4 = *(const v4f*)&s_B[buf][w][o];
    const float dv[TPL] = {d4.x, d4.y, d4.z, d4.w};
    const float uv[TPL] = {u4.x, u4.y, u4.z, u4.w};
    const float bv[TPL] = {b4.x, b4.y, b4.z, b4.w};

    float av[TPL], buv[TPL];
    #pragma unroll
    for (int i = 0; i < TPL; ++i) {
      av[i]  = __expf(dv[i] * A_dn);   // decay
      buv[i] = dv[i] * uv[i] * bv[i];  // input injection
    }

    // local inclusive compose of the lane's 4 steps -> lane aggregate (aa,bb)
    float aa = av[0], bb = buv[0];
    #pragma unroll
    for (int i = 1; i < TPL; ++i) {
      bb = fmaf(av[i], bb, buv[i]);
      aa = aa * av[i];
    }

    // Hillis-Steele inclusive scan of lane aggregates across 32 lanes:
    // combine(prev, cur) = (a_cur*a_prev, a_cur*b_prev + b_cur)
    #pragma unroll
    for (int off = 1; off < 32; off <<= 1) {
      const float ap = __shfl_up(aa, off, 32);
      const float bp = __shfl_up(bb, off, 32);
      if (t >= off) {
        bb = fmaf(aa, bp, bb);  // uses pre-update aa
        aa = aa * ap;
      }
    }

    // state at END of this lane's 4-step segment, given chunk carry h
    const float h_end = fmaf(aa, h, bb);
    // carry-in for this lane = previous lane's h_end (lane 0: chunk carry)
    float h_in = __shfl_up(h_end, 1, 32);
    if (t == 0) h_in = h;

    // regenerate the 4 states serially and emit one 16B NT store per lane
    v4f r;
    float hc = h_in;
    hc = fmaf(av[0], hc, buv[0]); r.x = hc;
    hc = fmaf(av[1], hc, buv[1]); r.y = hc;
    hc = fmaf(av[2], hc, buv[2]); r.z = hc;
    hc = fmaf(av[3], hc, buv[3]); r.w = hc;
    __builtin_nontemporal_store(r, (v4f*)(out_base + c * CH + o));

    h = __shfl(h_end, 31, 32);  // carry = state at end of chunk
  }
}

extern "C" void kernel_launch(void* const* d_in, const int* in_sizes, int n_in,
                              void* d_out, int out_size, void* d_ws, size_t ws_size,
                              hipStream_t stream) {
  // setup_inputs order: u, delta, A, B  (all float32)
  const float* u     = (const float*)d_in[0];
  const float* delta = (const float*)d_in[1];
  const float* A     = (const float*)d_in[2];
  const float* Bm    = (const float*)d_in[3];
  float* out = (float*)d_out;

  dim3 grid(BQ * DM);   // 2048 blocks, one per (b, d)
  dim3 block(NS * 32);  // 16 waves (one per state n), wave32
  SelectiveScanState_kernel<<<grid, block, 0, stream>>>(u, delta, A, Bm, out);
}